// GCN_68805376082492
// MI455X (gfx1250) — compile-verified
//
#include <hip/hip_runtime.h>
#include <hip/hip_bf16.h>
#include <math.h>

typedef __attribute__((ext_vector_type(2))) float v2f;
typedef __attribute__((ext_vector_type(8))) float v8f;

#define N_FEAT 512
#define HIDDEN 16
#define N_CLASSES 40

// ---------------------------------------------------------------------------
// Degree accumulation: one thread per edge, float atomics into L2-resident arrays
// ---------------------------------------------------------------------------
__global__ void degree_kernel(const int* __restrict__ src, const int* __restrict__ dst,
                              float* __restrict__ deg_out, float* __restrict__ deg_in, int e) {
    int i = blockIdx.x * blockDim.x + threadIdx.x;
    if (i >= e) return;
    atomicAdd(&deg_out[src[i]], 1.0f);
    atomicAdd(&deg_in[dst[i]], 1.0f);
}

// In-place deg -> rsqrt(max(deg,1)), over both arrays at once (2n contiguous)
__global__ void rsqrt_kernel(float* __restrict__ d, int total) {
    int i = blockIdx.x * blockDim.x + threadIdx.x;
    if (i >= total) return;
    float v = d[i];
    d[i] = rsqrtf(v < 1.0f ? 1.0f : v);
}

// ---------------------------------------------------------------------------
// GEMM1: H1[n][16] = (feat[n][512] * inv_out[n][None]) @ W1[512][16]
// (scaling folded into the A operand: exactly (feat @ W1) * inv_out per row)
//
// One wave computes TWO 16(node) x 16(hidden) tiles (32 rows) per K-loop,
// sharing the B fragment between both V_WMMA_F32_16X16X4_F32 issues.
//
// f32 16x16x4 lane layout (ISA 7.12.2):
//   A: lane L -> row M=L%16, VGPR0=K(2*(L/16)), VGPR1=K(2*(L/16)+1)  (contig float2)
//   B: lane L -> col N=L%16, same K pair (mirror of A)
//   C/D: VGPR r -> row r + 8*(L/16), col L%16
// ---------------------------------------------------------------------------
__global__ void gemm1_kernel(const float* __restrict__ feat,
                             const float* __restrict__ W1,
                             const float* __restrict__ inv_out,
                             float* __restrict__ h1, int n) {
    int wave = (int)((blockIdx.x * (unsigned)blockDim.x + threadIdx.x) >> 5);
    int lane = threadIdx.x & 31;
    int row0 = wave * 32;
    if (row0 >= n) return;                 // whole-wave guard: EXEC stays all-ones

    int half = lane >> 4;                  // 0 or 1
    int l16  = lane & 15;
    int koff = half * 2;

    int arow0 = row0 + l16;
    int arow1 = arow0 + 16;
    int ca0 = arow0 < n ? arow0 : n - 1;   // clamp only matters when n%32 != 0
    int ca1 = arow1 < n ? arow1 : n - 1;

    float inv0 = inv_out[ca0];
    float inv1 = inv_out[ca1];

    const float* ap0 = feat + (size_t)ca0 * N_FEAT + koff;
    const float* ap1 = feat + (size_t)ca1 * N_FEAT + koff;
    const float* bp  = W1 + l16;           // W1 row-major [512][16]

    v8f acc0 = {};
    v8f acc1 = {};
#pragma unroll 4
    for (int kb = 0; kb < N_FEAT; kb += 4) {
        v2f b, a0, a1;
        b.x  = bp[(kb + koff) * HIDDEN];
        b.y  = bp[(kb + koff + 1) * HIDDEN];
        a0.x = ap0[kb] * inv0;
        a0.y = ap0[kb + 1] * inv0;
        a1.x = ap1[kb] * inv1;
        a1.y = ap1[kb + 1] * inv1;
        acc0 = __builtin_amdgcn_wmma_f32_16x16x4_f32(
            false, a0, false, b, (short)0, acc0, false, false);
        acc1 = __builtin_amdgcn_wmma_f32_16x16x4_f32(
            false, a1, false, b, (short)0, acc1, false, false);
    }

    if (row0 + 32 <= n) {
        // Fast path (always taken for n % 32 == 0): branch-free clause of stores.
        float* p0 = h1 + (size_t)(row0 + half * 8) * HIDDEN + l16;
        float* p1 = p0 + 16 * HIDDEN;
#pragma unroll
        for (int r = 0; r < 8; ++r) {
            p0[r * HIDDEN] = acc0[r];
            p1[r * HIDDEN] = acc1[r];
        }
    } else {
#pragma unroll
        for (int r = 0; r < 8; ++r) {
            int rr0 = row0 + r + half * 8;
            int rr1 = rr0 + 16;
            if (rr0 < n) h1[(size_t)rr0 * HIDDEN + l16] = acc0[r];
            if (rr1 < n) h1[(size_t)rr1 * HIDDEN + l16] = acc1[r];
        }
    }
}

// ---------------------------------------------------------------------------
// Edge scatter-sum: 16 threads per edge, coalesced 64B gathers from the
// L2-resident [n][16] feature array, f32 atomics into the accumulator.
// ---------------------------------------------------------------------------
__global__ void agg_kernel(const float* __restrict__ hin,
                           const int* __restrict__ src, const int* __restrict__ dst,
                           float* __restrict__ agg, long long total) {
    long long t = (long long)blockIdx.x * blockDim.x + threadIdx.x;
    if (t >= total) return;
    int edge = (int)(t >> 4);
    int j    = (int)(t & 15);
    int s = src[edge];
    int d = dst[edge];
    atomicAdd(&agg[(size_t)d * HIDDEN + j], hin[(size_t)s * HIDDEN + j]);
}

// ---------------------------------------------------------------------------
// Between layers: h = relu(agg * inv_in + b1); h2pre = h * inv_out
// ---------------------------------------------------------------------------
__global__ void mid_kernel(const float* __restrict__ agg,
                           const float* __restrict__ inv_in,
                           const float* __restrict__ inv_out,
                           const float* __restrict__ b1,
                           float* __restrict__ h2, int n) {
    int t = blockIdx.x * blockDim.x + threadIdx.x;
    if (t >= n * HIDDEN) return;
    int node = t >> 4;
    int j    = t & 15;
    float v = agg[t] * inv_in[node] + b1[j];
    v = v > 0.0f ? v : 0.0f;
    h2[t] = v * inv_out[node];
}

// ---------------------------------------------------------------------------
// Final: logits = (agg * inv_in) @ W2 + b2, then log_softmax over 40 classes.
// Thread per node; W2 (2.5 KB) + b2 staged in LDS.
// ---------------------------------------------------------------------------
__global__ void final_kernel(const float* __restrict__ agg,
                             const float* __restrict__ inv_in,
                             const float* __restrict__ W2,
                             const float* __restrict__ b2,
                             float* __restrict__ out, int n) {
    __shared__ float sW2[HIDDEN * N_CLASSES];
    __shared__ float sb2[N_CLASSES];
    for (int i = threadIdx.x; i < HIDDEN * N_CLASSES; i += blockDim.x) sW2[i] = W2[i];
    for (int i = threadIdx.x; i < N_CLASSES; i += blockDim.x) sb2[i] = b2[i];
    __syncthreads();

    int nd = blockIdx.x * blockDim.x + threadIdx.x;
    if (nd >= n) return;

    float inv = inv_in[nd];
    float v[HIDDEN];
#pragma unroll
    for (int j = 0; j < HIDDEN; ++j) v[j] = agg[(size_t)nd * HIDDEN + j] * inv;

    float logit[N_CLASSES];
#pragma unroll
    for (int c = 0; c < N_CLASSES; ++c) {
        float a = sb2[c];
#pragma unroll
        for (int j = 0; j < HIDDEN; ++j) a = fmaf(v[j], sW2[j * N_CLASSES + c], a);
        logit[c] = a;
    }

    float mx = logit[0];
#pragma unroll
    for (int c = 1; c < N_CLASSES; ++c) mx = fmaxf(mx, logit[c]);
    float s = 0.0f;
#pragma unroll
    for (int c = 0; c < N_CLASSES; ++c) s += __expf(logit[c] - mx);
    float lse = mx + __logf(s);
#pragma unroll
    for (int c = 0; c < N_CLASSES; ++c)
        out[(size_t)nd * N_CLASSES + c] = logit[c] - lse;
}

// ---------------------------------------------------------------------------
extern "C" void kernel_launch(void* const* d_in, const int* in_sizes, int n_in,
                              void* d_out, int out_size, void* d_ws, size_t ws_size,
                              hipStream_t stream) {
    const float* feat = (const float*)d_in[0];
    const int*   src  = (const int*)d_in[1];
    const int*   dst  = (const int*)d_in[2];
    const float* W1   = (const float*)d_in[3];
    const float* b1   = (const float*)d_in[4];
    const float* W2   = (const float*)d_in[5];
    const float* b2   = (const float*)d_in[6];
    float*       out  = (float*)d_out;

    const int n = in_sizes[0] / N_FEAT;   // 100000
    const int e = in_sizes[1];            // 3200000

    // Workspace layout (floats): deg_out[n] | deg_in[n] | bufA[n*16] | bufB[n*16]
    float* deg_out = (float*)d_ws;
    float* deg_in  = deg_out + n;
    float* bufA    = deg_in + n;                 // H1pre / H2pre
    float* bufB    = bufA + (size_t)n * HIDDEN;  // scatter accumulator

    const int T = 256;

    // Degrees -> inv-sqrt (in place)
    hipMemsetAsync(deg_out, 0, 2 * (size_t)n * sizeof(float), stream);
    degree_kernel<<<(e + T - 1) / T, T, 0, stream>>>(src, dst, deg_out, deg_in, e);
    rsqrt_kernel<<<(2 * n + T - 1) / T, T, 0, stream>>>(deg_out, 2 * n);

    // Layer 1 GEMM (WMMA f32, 2 M-tiles per wave) fused with src-side norm
    int waves  = (n + 31) / 32;
    int blocks = (waves + (T / 32) - 1) / (T / 32);
    gemm1_kernel<<<blocks, T, 0, stream>>>(feat, W1, deg_out, bufA, n);

    // Layer 1 aggregation
    long long tot = (long long)e * HIDDEN;
    hipMemsetAsync(bufB, 0, (size_t)n * HIDDEN * sizeof(float), stream);
    agg_kernel<<<(int)((tot + T - 1) / T), T, 0, stream>>>(bufA, src, dst, bufB, tot);

    // inv_in scale + bias + ReLU + layer-2 src-side norm
    mid_kernel<<<(n * HIDDEN + T - 1) / T, T, 0, stream>>>(bufB, deg_in, deg_out, b1, bufA, n);

    // Layer 2 aggregation
    hipMemsetAsync(bufB, 0, (size_t)n * HIDDEN * sizeof(float), stream);
    agg_kernel<<<(int)((tot + T - 1) / T), T, 0, stream>>>(bufA, src, dst, bufB, tot);

    // inv_in scale + W2 + b2 + log_softmax
    final_kernel<<<(n + T - 1) / T, T, 0, stream>>>(bufB, deg_in, W2, b2, out, n);
}